// QuantumLayer_39891656245782
// MI455X (gfx1250) — compile-verified
//
#include <hip/hip_runtime.h>
#include <hip/hip_bf16.h>

// ---------------------------------------------------------------------------
// 10-qubit variational circuit simulator for MI455X (gfx1250, wave32).
//  * one wave == one batch element; state = 64 f32 VGPRs (32 complex / lane)
//  * amplitude index a: bits 0..4 = lane (qubits 0..4), bits 5..9 = reg (q 5..9)
//  * lane-qubit gates via __shfl_xor (ds_swizzle), reg-qubit gates pure VALU
//  * <Z_i> readout for qubits 5..9 via 16 chained v_wmma_f32_16x16x4_f32
//    (exact f32 contraction against a +-1 sign matrix); qubits 0..4 via
//    signed xor-butterfly reductions.
// ---------------------------------------------------------------------------

typedef __attribute__((ext_vector_type(2))) float v2f;
typedef __attribute__((ext_vector_type(8))) float v8f;

__device__ __forceinline__ float sxor(float v, int m) { return __shfl_xor(v, m, 32); }

// Generic 1-qubit complex 2x2 gate on qubit Q.
template<int Q>
__device__ __forceinline__ void apply1q(float (&sr)[32], float (&si)[32],
    float g00r, float g00i, float g01r, float g01i,
    float g10r, float g10i, float g11r, float g11i, int lane)
{
    if constexpr (Q >= 5) {
        constexpr int m = 1 << (Q - 5);           // register-bit qubit
#pragma unroll
        for (int r = 0; r < 32; ++r) {
            if (r & m) continue;
            const int r1 = r | m;
            float ar = sr[r],  ai = si[r];
            float br = sr[r1], bi = si[r1];
            sr[r]  = g00r*ar - g00i*ai + g01r*br - g01i*bi;
            si[r]  = g00r*ai + g00i*ar + g01r*bi + g01i*br;
            sr[r1] = g10r*ar - g10i*ai + g11r*br - g11i*bi;
            si[r1] = g10r*ai + g10i*ar + g11r*bi + g11i*br;
        }
    } else {
        constexpr int m = 1 << Q;                 // lane-bit qubit
        const bool hb = (lane & m) != 0;
        const float ar_ = hb ? g11r : g00r;       // coeff for own amp
        const float ai_ = hb ? g11i : g00i;
        const float br_ = hb ? g10r : g01r;       // coeff for partner amp
        const float bi_ = hb ? g10i : g01i;
#pragma unroll
        for (int r = 0; r < 32; ++r) {
            const float o_r = sxor(sr[r], m);
            const float o_i = sxor(si[r], m);
            const float a = sr[r], b = si[r];
            sr[r] = ar_*a - ai_*b + br_*o_r - bi_*o_i;
            si[r] = ar_*b + ai_*a + br_*o_i + bi_*o_r;
        }
    }
}

// Entangler on (J, (J+1)%10): even J -> IsingXX, odd J -> IsingZZ.
template<int J>
__device__ __forceinline__ void ising(float (&sr)[32], float (&si)[32],
                                      float c, float s, int lane)
{
    if constexpr ((J & 1) == 0) {
        // IsingXX: psi' = c*psi - i*s*(X x X)psi ; partner = a XOR mask
        if constexpr (J == 0 || J == 2) {
            constexpr int lm = 3 << J;            // both bits in lane
#pragma unroll
            for (int r = 0; r < 32; ++r) {
                const float orr = sxor(sr[r], lm);
                const float oii = sxor(si[r], lm);
                const float a = sr[r], b = si[r];
                sr[r] = c*a + s*oii;
                si[r] = c*b - s*orr;
            }
        } else if constexpr (J == 4) {
            // lane bit 4 + reg bit 0 ; all reads before writes per pair
#pragma unroll
            for (int r = 0; r < 32; r += 2) {
                const float o0r = sxor(sr[r + 1], 16);
                const float o0i = sxor(si[r + 1], 16);
                const float o1r = sxor(sr[r],     16);
                const float o1i = sxor(si[r],     16);
                const float a0 = sr[r],   b0 = si[r];
                const float a1 = sr[r+1], b1 = si[r+1];
                sr[r]   = c*a0 + s*o0i;  si[r]   = c*b0 - s*o0r;
                sr[r+1] = c*a1 + s*o1i;  si[r+1] = c*b1 - s*o1r;
            }
        } else {
            constexpr int rm  = (J == 6) ? 6  : 24;   // reg-bit pair mask
            constexpr int top = (J == 6) ? 4  : 16;   // MSB of mask
#pragma unroll
            for (int r = 0; r < 32; ++r) {
                if (r & top) continue;
                const int r2 = r ^ rm;
                const float a0 = sr[r],  b0 = si[r];
                const float a1 = sr[r2], b1 = si[r2];
                sr[r]  = c*a0 + s*b1;  si[r]  = c*b0 - s*a1;
                sr[r2] = c*a1 + s*b0;  si[r2] = c*b1 - s*a0;
            }
        }
    } else {
        // IsingZZ: multiply by (c - i*sign*s), sign=+1 iff bits equal
        if constexpr (J == 1 || J == 3) {
            const bool eq = (((lane >> J) & 1) == ((lane >> (J + 1)) & 1));
            const float ss = eq ? s : -s;
#pragma unroll
            for (int r = 0; r < 32; ++r) {
                const float a = sr[r], b = si[r];
                sr[r] = c*a + ss*b;
                si[r] = c*b - ss*a;
            }
        } else if constexpr (J == 5 || J == 7) {
            constexpr int b0 = J - 5, b1 = J - 4;     // reg bits
#pragma unroll
            for (int r = 0; r < 32; ++r) {
                const float ss = (((r >> b0) & 1) == ((r >> b1) & 1)) ? s : -s;
                const float a = sr[r], b = si[r];
                sr[r] = c*a + ss*b;
                si[r] = c*b - ss*a;
            }
        } else {  // J == 9 : reg bit 4 with lane bit 0
            const float ssL = (lane & 1) ? -s : s;
#pragma unroll
            for (int r = 0; r < 32; ++r) {
                const float ss = (r & 16) ? -ssL : ssL;
                const float a = sr[r], b = si[r];
                sr[r] = c*a + ss*b;
                si[r] = c*b - ss*a;
            }
        }
    }
}

// Apply the (layer-independent) batched Rot gate on every qubit.
template<int J>
__device__ __forceinline__ void rot_all(float (&sr)[32], float (&si)[32],
                                        const float (&g)[80], int lane)
{
    apply1q<J>(sr, si, g[8*J+0], g[8*J+1], g[8*J+2], g[8*J+3],
                        g[8*J+4], g[8*J+5], g[8*J+6], g[8*J+7], lane);
    if constexpr (J < 9) rot_all<J + 1>(sr, si, g, lane);
}

// Fixed Rx/Ry rotation then entangler, for each qubit in order.
template<int J>
__device__ __forceinline__ void fe_all(float (&sr)[32], float (&si)[32],
                                       int lane, int layer,
                                       const float* __restrict__ rotw,
                                       const float* __restrict__ entw)
{
    const float th = rotw[J * 4 + layer];          // uniform -> s_load
    const float c  = __cosf(0.5f * th);
    const float s  = __sinf(0.5f * th);
    const bool isRx = (((layer + J) & 1) == 0);
    // Rx: [[c,-is],[-is,c]]   Ry: [[c,-s],[s,c]]
    const float g01r = isRx ? 0.f : -s, g01i = isRx ? -s : 0.f;
    const float g10r = isRx ? 0.f :  s, g10i = isRx ? -s : 0.f;
    apply1q<J>(sr, si, c, 0.f, g01r, g01i, g10r, g10i, c, 0.f, lane);

    const float te = entw[J * 4 + layer];
    const float ce = __cosf(0.5f * te);
    const float se = __sinf(0.5f * te);
    ising<J>(sr, si, ce, se, lane);
    if constexpr (J < 9) fe_all<J + 1>(sr, si, lane, layer, rotw, entw);
}

__global__ __launch_bounds__(256)
void qsim10_kernel(const float* __restrict__ x,      // (B, 30)
                   const float* __restrict__ rotw,   // (10, 4)
                   const float* __restrict__ entw,   // (10, 4)
                   float* __restrict__ out,          // (B, 10)
                   int B)
{
    const int lane = threadIdx.x & 31;
    const int b    = blockIdx.x * 8 + (threadIdx.x >> 5);
    if (b >= B) return;

    // --- per-batch Rot gate table (identical across layers) ---------------
    const float* xb = x + b * 30;
    float g[80];
#pragma unroll
    for (int j = 0; j < 10; ++j) {
        const float phi = xb[3 * j + 0];
        const float th  = xb[3 * j + 1];
        const float om  = xb[3 * j + 2];
        const float ct = __cosf(0.5f * th), st = __sinf(0.5f * th);
        const float ap = 0.5f * (phi + om), am = 0.5f * (phi - om);
        const float cp = __cosf(ap), sp = __sinf(ap);
        const float cm = __cosf(am), sm = __sinf(am);
        g[8*j+0] =  cp * ct;  g[8*j+1] = -sp * ct;   // g00 = e^{-i(phi+om)/2} c
        g[8*j+2] = -cm * st;  g[8*j+3] = -sm * st;   // g01 = -e^{ i(phi-om)/2} s
        g[8*j+4] =  cm * st;  g[8*j+5] = -sm * st;   // g10 =  e^{-i(phi-om)/2} s
        g[8*j+6] =  cp * ct;  g[8*j+7] =  sp * ct;   // g11 = e^{ i(phi+om)/2} c
    }

    // --- state |0...0> -----------------------------------------------------
    float sr[32], si[32];
#pragma unroll
    for (int r = 0; r < 32; ++r) { sr[r] = 0.f; si[r] = 0.f; }
    if (lane == 0) sr[0] = 1.f;

    // --- 4 layers ----------------------------------------------------------
    for (int layer = 0; layer < 4; ++layer) {
        rot_all<0>(sr, si, g, lane);
        fe_all<0>(sr, si, lane, layer, rotw, entw);
    }

    // --- probabilities -----------------------------------------------------
    float p[32];
#pragma unroll
    for (int r = 0; r < 32; ++r) p[r] = sr[r] * sr[r] + si[r] * si[r];

    // per-lane total probability (for lane-bit qubits 0..4)
    float u = 0.f;
#pragma unroll
    for (int r = 0; r < 32; ++r) u += p[r];

    // qubits 0..4: signed xor-butterfly (exact)
    float evlow[5];
#pragma unroll
    for (int i = 0; i < 5; ++i) {
        float w = ((lane >> i) & 1) ? -u : u;
        w += sxor(w, 1); w += sxor(w, 2); w += sxor(w, 4);
        w += sxor(w, 8); w += sxor(w, 16);
        evlow[i] = w;
    }

    // --- qubits 5..9 via v_wmma_f32_16x16x4_f32 ---------------------------
    // A 16x4 layout: lanes 0-15 hold K=0 (vgpr0)/K=1 (vgpr1); lanes 16-31
    // hold K=2/K=3.  B 4x16: row K striped across lanes (K tied to lane-half
    // + vgpr like A), N = lane%16.  Two passes x 8 K-chunks cover every
    // (lane, reg) probability exactly once, each multiplied by the exact
    // +-1.0f sign of reg-bit n (qubit 5+n), accumulated in f32.
    const int n  = lane & 15;
    const int hi = (lane >> 4) & 1;
    v8f acc = {0.f, 0.f, 0.f, 0.f, 0.f, 0.f, 0.f, 0.f};
#pragma unroll
    for (int pass = 0; pass < 2; ++pass) {
#pragma unroll
        for (int cch = 0; cch < 8; ++cch) {
            const bool sel = (pass == 0) ? (hi != 0) : (hi == 0);
            const float ax = sel ? p[4 * cch + 2] : p[4 * cch + 0];
            const float ay = sel ? p[4 * cch + 3] : p[4 * cch + 1];
            const int k0 = 2 * hi;
            const int r0 = 4 * cch + ((pass == 0) ? k0 : (2 - k0));
            const int r1 = r0 + 1;
            const float bx = (n < 5) ? (((r0 >> n) & 1) ? -1.f : 1.f) : 0.f;
            const float by = (n < 5) ? (((r1 >> n) & 1) ? -1.f : 1.f) : 0.f;
            v2f A; A.x = ax; A.y = ay;
            v2f Bm; Bm.x = bx; Bm.y = by;
            acc = __builtin_amdgcn_wmma_f32_16x16x4_f32(
                false, A, false, Bm, (short)0, acc, false, false);
        }
    }
    // D[m,n]: lane holds column n, rows m = v + 8*hi -> sum 8 vgprs + halves
    float sh = acc[0] + acc[1] + acc[2] + acc[3] +
               acc[4] + acc[5] + acc[6] + acc[7];
    sh += sxor(sh, 16);

    // --- write (B,10) ------------------------------------------------------
    float* ob = out + b * 10;
    if (lane < 5) {
        ob[5 + lane] = sh;                       // qubits 5..9 (column = lane)
    } else if (lane < 10) {
        const int i = lane - 5;
        const float v = (i == 0) ? evlow[0] : (i == 1) ? evlow[1] :
                        (i == 2) ? evlow[2] : (i == 3) ? evlow[3] : evlow[4];
        ob[i] = v;                               // qubits 0..4
    }
}

extern "C" void kernel_launch(void* const* d_in, const int* in_sizes, int n_in,
                              void* d_out, int out_size, void* d_ws, size_t ws_size,
                              hipStream_t stream) {
    const float* x    = (const float*)d_in[0];   // (B, 30) float32
    const float* rotw = (const float*)d_in[1];   // (10, 4) float32
    const float* entw = (const float*)d_in[2];   // (10, 4) float32
    float* out        = (float*)d_out;           // (B, 10) float32
    const int B = in_sizes[0] / 30;
    const int blocks = (B + 7) / 8;              // 8 waves (batch elems) / block
    qsim10_kernel<<<blocks, 256, 0, stream>>>(x, rotw, entw, out, B);
}